// MaxAbsPool2D_10050223472947
// MI455X (gfx1250) — compile-verified
//
#include <hip/hip_runtime.h>
#include <hip/hip_bf16.h>

// MaxAbs 2x2 pooling, NHWC fp32: in (B,224,224,96) -> out (B,112,112,96).
// 2D grid: blockIdx.y = output row (b*Ho+ho), 12 blocks of 224 threads cover
// one row's 2688 float4 outputs exactly (no bounds check, no idle lanes).
// 4x b128 NT loads + 1x b128 NT store per thread; strict-> abs-compare chain
// in window order to match jnp.argmax first-occurrence tie semantics.

typedef float v4f __attribute__((ext_vector_type(4)));

__global__ __launch_bounds__(224) void MaxAbsPool2D_kernel(
    const float* __restrict__ x, float* __restrict__ out)
{
    constexpr int C   = 96;
    constexpr int CV  = C / 4;        // 24 float4 chunks per pixel
    constexpr int W   = 224;
    constexpr int RWV = 112 * CV;     // 2688 float4s per output row

    // float4 index within this output row: 0..2687
    const int tid = blockIdx.x * 224 + threadIdx.x;
    const int row = blockIdx.y;       // b*Ho + ho  (input row pair starts at 2*row)

    const int wo = tid / CV;          // output column (single small-range div)
    const int cv = tid - wo * CV;     // channel chunk

    // Scalar (SALU) part: start of the input row pair for this output row.
    const long rowbase = (long)row * (2L * W * C);       // (2*row)*W*C
    // Per-lane part: window top-left, this channel chunk.
    const long base = rowbase + (long)(2 * wo) * C + (long)cv * 4;

    const v4f* pa = (const v4f*)(x + base);                   // (dy=0, dx=0)
    const v4f* pb = (const v4f*)(x + base + C);               // (dy=0, dx=1)
    const v4f* pc = (const v4f*)(x + base + (long)W * C);     // (dy=1, dx=0)
    const v4f* pd = (const v4f*)(x + base + (long)W * C + C); // (dy=1, dx=1)

    v4f a = __builtin_nontemporal_load(pa);
    v4f b = __builtin_nontemporal_load(pb);
    v4f c = __builtin_nontemporal_load(pc);
    v4f d = __builtin_nontemporal_load(pd);

    v4f r;
#pragma unroll
    for (int i = 0; i < 4; ++i) {
        float best  = a[i];
        float besta = __builtin_fabsf(a[i]);

        float vb = b[i], fb = __builtin_fabsf(b[i]);
        if (fb > besta) { best = vb; besta = fb; }

        float vc = c[i], fc = __builtin_fabsf(c[i]);
        if (fc > besta) { best = vc; besta = fc; }

        float vd = d[i], fd = __builtin_fabsf(d[i]);
        if (fd > besta) { best = vd; }

        r[i] = best;
    }

    __builtin_nontemporal_store(r, (v4f*)(out + (long)row * (RWV * 4) + (long)tid * 4));
}

extern "C" void kernel_launch(void* const* d_in, const int* in_sizes, int n_in,
                              void* d_out, int out_size, void* d_ws, size_t ws_size,
                              hipStream_t stream)
{
    const float* x = (const float*)d_in[0];
    float* out     = (float*)d_out;

    // B derived from input size; H=W=224, C=96 fixed by the reference.
    const int B  = in_sizes[0] / (224 * 224 * 96);
    const int Ho = 112;

    // grid.x: 12 blocks x 224 threads = 2688 float4s = one output row, exact.
    // grid.y: one output row (b*Ho + ho) per y-slice.
    dim3 grid(12, B * Ho, 1);
    dim3 block(224, 1, 1);
    MaxAbsPool2D_kernel<<<grid, block, 0, stream>>>(x, out);
}